// MeanLastStateBertNERwithoutBERTtags_79894981640528
// MI455X (gfx1250) — compile-verified
//
#include <hip/hip_runtime.h>

typedef __attribute__((ext_vector_type(2))) float v2f;
typedef __attribute__((ext_vector_type(8))) float v8f;

#define VOCAB 30522
#define DHID  768
#define NCOL  9
#define KSTEPS (DHID / 4)   // 192 WMMA k-steps of K=4

// ---------------- Kernel 0: zero segment accumulators, compute Wsum2 ----------------
__global__ void prep_kernel(const float* __restrict__ W,
                            float* __restrict__ seg_sum,   // [VOCAB] (seg_cnt contiguous after)
                            float* __restrict__ wsum2) {   // [16]
    int gid = blockIdx.x * blockDim.x + threadIdx.x;
    if (gid < 2 * VOCAB) seg_sum[gid] = 0.0f;  // covers seg_sum + seg_cnt
    if (blockIdx.x == 0 && threadIdx.x < 16) {
        float s = 0.0f;
        if (threadIdx.x < NCOL) {
            for (int d = 0; d < DHID; ++d) s += W[(DHID + d) * NCOL + threadIdx.x];
        }
        wsum2[threadIdx.x] = s;   // cols 9..15 = 0
    }
}

// ---------------- Kernel 1: per-row mean + segment accumulation ----------------
__global__ void __launch_bounds__(256)
rowmean_kernel(const float* __restrict__ h, const int* __restrict__ ids,
               float* __restrict__ seg_sum, float* __restrict__ seg_cnt, int rows) {
    int wave = threadIdx.x >> 5;
    int lane = threadIdx.x & 31;
    int row  = blockIdx.x * (blockDim.x >> 5) + wave;
    if (row >= rows) return;                       // wave-uniform
    const float* hr = h + (size_t)row * DHID;
    float s = 0.0f;
    #pragma unroll
    for (int i = 0; i < DHID / 32; ++i) s += hr[lane + 32 * i];   // coalesced
    #pragma unroll
    for (int m = 16; m >= 1; m >>= 1) s += __shfl_xor(s, m, 32);
    if (lane == 0) {
        int id = ids[row];
        atomicAdd(&seg_sum[id], s * (1.0f / DHID));
        atomicAdd(&seg_cnt[id], 1.0f);
    }
}

// ---------------- Kernel 2: WMMA GEMM (16-row tile per wave) + rank-1 ctx + bias ----------------
__global__ void __launch_bounds__(256)
gemm_kernel(const float* __restrict__ h, const int* __restrict__ ids,
            const float* __restrict__ W, const float* __restrict__ bias,
            const float* __restrict__ seg_sum, const float* __restrict__ seg_cnt,
            const float* __restrict__ wsum2, float* __restrict__ out, int rows) {
    // B fragments pre-swizzled into WMMA lane layout:
    // step t, lane L holds W[4t + 2*(L>=16) + {0,1}, L&15] (zero-padded cols 9..15)
    __shared__ float bfrag[KSTEPS * 64];   // 48 KB
    for (int idx = threadIdx.x; idx < KSTEPS * 64; idx += blockDim.x) {
        int t    = idx >> 6;
        int r    = idx & 63;
        int lane = r >> 1;
        int j    = r & 1;
        int d    = 4 * t + ((lane >= 16) ? 2 : 0) + j;
        int c    = lane & 15;
        bfrag[idx] = (c < NCOL) ? W[d * NCOL + c] : 0.0f;
    }
    __syncthreads();

    int wave   = threadIdx.x >> 5;
    int lane   = threadIdx.x & 31;
    int tile   = blockIdx.x * (blockDim.x >> 5) + wave;
    int nTiles = rows >> 4;
    if (tile >= nTiles) return;                    // wave-uniform

    int rowBase = tile << 4;
    int rowA    = rowBase + (lane & 15);           // A-matrix M index for this lane
    int kOff    = (lane >= 16) ? 2 : 0;            // A-matrix K pair select
    const float* aptr = h + (size_t)rowA * DHID + kOff;

    v8f acc = {};
    #pragma unroll 4
    for (int t = 0; t < KSTEPS; ++t) {
        v2f a  = *(const v2f*)(aptr + 4 * t);                   // global_load_b64
        v2f bb = *(const v2f*)(&bfrag[t * 64 + lane * 2]);      // ds_load_b64, conflict-free
        acc = __builtin_amdgcn_wmma_f32_16x16x4_f32(
            /*neg_a=*/false, a, /*neg_b=*/false, bb,
            /*c_mod=*/(short)0, acc, /*reuse_a=*/false, /*reuse_b=*/false);
    }

    // C/D layout: VGPR r, lanes 0-15 -> M=r, lanes 16-31 -> M=r+8; N = lane&15
    int col = lane & 15;
    if (col < NCOL) {
        float ws = wsum2[col];
        float bc = bias[col];
        int   mh = (lane >= 16) ? 8 : 0;
        #pragma unroll
        for (int r = 0; r < 8; ++r) {
            int   row = rowBase + r + mh;
            int   id  = ids[row];
            float ctx = seg_sum[id] / fmaxf(seg_cnt[id], 1.0f);
            out[(size_t)row * NCOL + col] = acc[r] + ctx * ws + bc;
        }
    }
}

extern "C" void kernel_launch(void* const* d_in, const int* in_sizes, int n_in,
                              void* d_out, int out_size, void* d_ws, size_t ws_size,
                              hipStream_t stream) {
    const int*   ids  = (const int*)d_in[0];    // batch [32,512] int32
    const float* h    = (const float*)d_in[1];  // last_hidden_state [32,512,768] f32
    const float* W    = (const float*)d_in[2];  // [1536,9] f32
    const float* bias = (const float*)d_in[3];  // [9] f32
    float*       out  = (float*)d_out;          // [32,512,9] f32
    int rows = in_sizes[0];                     // 16384

    float* seg_sum = (float*)d_ws;              // [VOCAB]
    float* seg_cnt = seg_sum + VOCAB;           // [VOCAB]
    float* wsum2   = seg_cnt + VOCAB;           // [16]

    int zblocks = (2 * VOCAB + 255) / 256;
    prep_kernel<<<zblocks, 256, 0, stream>>>(W, seg_sum, wsum2);

    int waves_per_block = 256 / 32;
    int rblocks = (rows + waves_per_block - 1) / waves_per_block;
    rowmean_kernel<<<rblocks, 256, 0, stream>>>(h, ids, seg_sum, seg_cnt, rows);

    int nTiles = rows / 16;
    int gblocks = (nTiles + waves_per_block - 1) / waves_per_block;
    gemm_kernel<<<gblocks, 256, 0, stream>>>(h, ids, W, bias, seg_sum, seg_cnt, wsum2, out, rows);
}